// MeanAggregator_34557306863776
// MI455X (gfx1250) — compile-verified
//
#include <hip/hip_runtime.h>
#include <hip/hip_bf16.h>
#include <stdint.h>

// Mean neighbor aggregation (GraphSAGE-style), CDNA5 / gfx1250.
//
// One wave32 per output row b; a feature row (128 fp32 = 512 B) is exactly one
// b128 transfer across 32 lanes (16 B/lane).
//
// Data movement: always issue 10 async global->LDS row copies (ASYNCcnt DMA,
// no destination VGPR pressure), then reduce out of LDS with ds_load_b128,
// waiting s_wait_asynccnt (9-s) before slot s (async loads complete in order)
// so the reduction overlaps the tail of the DMA. Masking (s < n) and the
// 1/n scale are folded into per-slot FMA coefficients; cnt==0 falls back to
// the node's own feature row via slot 0.
//
// Scalar path: b is wave-uniform, so readfirstlane(b) lets the compiler emit
// SMEM s_load for count / own node / all 10 neighbor indices.

#define WAVES_PER_BLOCK 8
#define NUM_SAMPLE 10
#define D_FEAT 128

__global__ __launch_bounds__(WAVES_PER_BLOCK * 32)
void mean_agg_kernel(const int* __restrict__ nodes,
                     const int* __restrict__ neighbor_idx,
                     const int* __restrict__ neighbor_count,
                     const float* __restrict__ features,
                     float* __restrict__ out,
                     int B) {
    // 8 waves * 10 rows * 512 B = 40 KB LDS per block (320 KB/WGP -> 8 blocks/WGP)
    __shared__ __align__(16) float lds[WAVES_PER_BLOCK][NUM_SAMPLE][D_FEAT];

    const int wave = threadIdx.x >> 5;
    const int lane = threadIdx.x & 31;
    const int b = blockIdx.x * WAVES_PER_BLOCK + wave;
    if (b >= B) return;

    // b is uniform per wave: force scalar so index/count fetches become SMEM.
    const int bs = __builtin_amdgcn_readfirstlane(b);

    int cnt = neighbor_count[bs];                       // s_load
    cnt = (cnt < 0) ? 0 : (cnt > NUM_SAMPLE ? NUM_SAMPLE : cnt);
    const int n = (cnt == 0) ? 1 : cnt;

    int rows[NUM_SAMPLE];
    #pragma unroll
    for (int s = 0; s < NUM_SAMPLE; ++s)
        rows[s] = neighbor_idx[bs * NUM_SAMPLE + s];    // s_load_b256 + s_load_b64
    if (cnt == 0) rows[0] = nodes[bs];                  // own-feature fallback

    // Flat LDS addresses keep the aperture in bits [63:32]; low 32 bits are the
    // LDS byte offset the async instruction needs.
    const uint32_t lane_byte = (uint32_t)(lane << 4);   // 16 B per lane
    const uint32_t lds_base  = (uint32_t)(uintptr_t)&lds[wave][0][0] + lane_byte;
    const uint64_t feat_base = (uint64_t)(uintptr_t)features;

    // Issue all 10 row copies: global -> LDS, 512 B per instruction
    // (32 lanes x 16 B), tracked by ASYNCcnt.
    #pragma unroll
    for (int s = 0; s < NUM_SAMPLE; ++s) {
        const unsigned long long gaddr =
            feat_base + ((unsigned long long)(uint32_t)rows[s] << 9) + lane_byte;
        const uint32_t laddr = lds_base + (uint32_t)(s << 9);
        asm volatile("global_load_async_to_lds_b128 %0, %1, off"
                     :: "v"(laddr), "v"(gaddr)
                     : "memory");
    }

    // Masked mean: coefficient inv for live slots, 0 for padding slots.
    const float inv = 1.0f / (float)n;
    float4 acc = make_float4(0.f, 0.f, 0.f, 0.f);
    #pragma unroll
    for (int s = 0; s < NUM_SAMPLE; ++s) {
        // Async loads complete in order: slot s landed once ASYNCcnt <= 9-s.
        asm volatile("s_wait_asynccnt %0" :: "i"(NUM_SAMPLE - 1 - s) : "memory");
        const float m = (s < n) ? inv : 0.0f;
        const float4 v = *(const float4*)&lds[wave][s][lane << 2];  // ds_load_b128
        acc.x = fmaf(m, v.x, acc.x);
        acc.y = fmaf(m, v.y, acc.y);
        acc.z = fmaf(m, v.z, acc.z);
        acc.w = fmaf(m, v.w, acc.w);
    }

    *(float4*)&out[(size_t)b * D_FEAT + (lane << 2)] = acc;         // global_store_b128
}

extern "C" void kernel_launch(void* const* d_in, const int* in_sizes, int n_in,
                              void* d_out, int out_size, void* d_ws, size_t ws_size,
                              hipStream_t stream) {
    const int*   nodes          = (const int*)d_in[0];   // [B] int32
    const int*   neighbor_idx   = (const int*)d_in[1];   // [B,10] int32
    const int*   neighbor_count = (const int*)d_in[2];   // [B] int32
    const float* features       = (const float*)d_in[3]; // [N,128] fp32
    float*       out            = (float*)d_out;         // [B,128] fp32

    const int B = in_sizes[0];
    const int blocks = (B + WAVES_PER_BLOCK - 1) / WAVES_PER_BLOCK;
    mean_agg_kernel<<<blocks, WAVES_PER_BLOCK * 32, 0, stream>>>(
        nodes, neighbor_idx, neighbor_count, features, out, B);
}